// DepthLossV2_43696997269957
// MI455X (gfx1250) — compile-verified
//
#include <hip/hip_runtime.h>
#include <hip/hip_bf16.h>

typedef __attribute__((ext_vector_type(2))) float v2f;
typedef __attribute__((ext_vector_type(8))) float v8f;

#define TILE     128
#define NTHREADS 256
#define NWAVES   (NTHREADS / 32)

// Wave-wide (32-lane) sum using the CDNA5 f32 WMMA unit.
// A (16x4) holds the 32 lane accumulators in column 0 (lanes 0-15) and
// column 2 (lanes 16-31); B = ones(4x16) -> D[m][n] = acc_m + acc_{m+16}.
// Each lane sums its 8 D VGPRs, then one xor-16 shuffle-add folds the two
// half-waves: every lane ends with the full wave sum.
__device__ __forceinline__ float wave_sum_wmma(float acc) {
#if __has_builtin(__builtin_amdgcn_wmma_f32_16x16x4_f32)
    v2f a; a[0] = acc;  a[1] = 0.0f;
    v2f b; b[0] = 1.0f; b[1] = 1.0f;
    v8f c = {};
    v8f d = __builtin_amdgcn_wmma_f32_16x16x4_f32(
        /*neg_a=*/false, a, /*neg_b=*/false, b,
        /*c_mod=*/(short)0, c, /*reuse_a=*/false, /*reuse_b=*/false);
    float s = ((d[0] + d[1]) + (d[2] + d[3])) + ((d[4] + d[5]) + (d[6] + d[7]));
    s += __shfl_xor(s, 16, 32);
    return s;
#else
    #pragma unroll
    for (int off = 16; off > 0; off >>= 1) acc += __shfl_xor(acc, off, 32);
    return acc;
#endif
}

// Inner 64-column strip. MASKED=false is the hot path (full off-diagonal
// tile): no per-pair masking at all -> ~9 dual-issuable VALU ops per pair.
// MASKED=true handles diagonal / ragged tiles with a single k<=kmax test.
template <bool MASKED>
__device__ __forceinline__ float strip_accumulate(const float* __restrict__ jt,
                                                  float pi, float dif0, int kmax,
                                                  float c1, float c2) {
    float acc = 0.0f;
    float dif = dif0;
    #pragma unroll 8
    for (int k = 0; k < 64; ++k) {
        const float pj = jt[k];
        float d  = pi - pj;
        float d1 = fmaf(-c1, dif, d);               // d - 0.2*steps
        float d2 = fmaxf(fmaf(-c2, dif, d1), 0.0f); // max(d1 - 0.8*steps, 0)
        float rv = (d >= 0.0f) ? ((d1 >= 0.0f) ? d2 : d1) : d;
        if (MASKED) acc += (k <= kmax) ? fabsf(rv) : 0.0f;
        else        acc += fabsf(rv);               // abs = free src modifier
        dif -= 1.0f;
    }
    return acc;
}

// One block per lower-triangular 128x128 tile (ti >= tj).
// 256 threads: 2 threads per row, 64 columns each, operands staged in LDS.
__global__ __launch_bounds__(NTHREADS)
void depth_loss_tile_kernel(const float* __restrict__ p,
                            const float* __restrict__ z_spacing,
                            const float* __restrict__ nth_slice,
                            float* __restrict__ block_sums,
                            int n) {
    __shared__ float jtile[TILE];
    __shared__ float itile[TILE];
    __shared__ float wsum[NWAVES];

    const float s  = 1.0f * z_spacing[0] * nth_slice[0]; // STEP * zsp * nth
    const float c1 = 0.2f * s;
    const float c2 = 0.8f * s;

    // Decode linear block id -> (ti, tj) with ti >= tj, b = ti*(ti+1)/2 + tj.
    int b  = (int)blockIdx.x;
    int ti = (int)((sqrtf(8.0f * (float)b + 1.0f) - 1.0f) * 0.5f);
    while ((ti + 1) * (ti + 2) / 2 <= b) ti++;
    while (ti * (ti + 1) / 2 > b)        ti--;
    int tj = b - ti * (ti + 1) / 2;

    const int i0 = ti * TILE;
    const int j0 = tj * TILE;
    const int t  = (int)threadIdx.x;

    // Cooperative staging: threads 0-127 load the j tile, 128-255 the i tile.
    if (t < TILE) {
        int j = j0 + t;
        jtile[t] = (j < n) ? p[j] : 0.0f;
    } else {
        int i = i0 + (t - TILE);
        itile[t - TILE] = (i < n) ? p[i] : 0.0f;
    }
    __syncthreads();

    const int   r     = t >> 1;          // local row 0..127
    const int   half  = t & 1;           // which 64-column half
    const int   i     = i0 + r;
    const int   jbase = half * 64;
    const float pi    = itile[r];
    const float dif0  = (float)(i - (j0 + jbase)); // i - j at k = 0

    // Block-uniform specialization: keeps EXEC all-1s for the WMMA below.
    const bool fullTile = (i0 + TILE <= n) && (j0 + TILE <= n);

    float acc;
    if ((ti != tj) && fullTile) {
        // Hot path (97% of blocks): i > j and in-bounds guaranteed.
        acc = strip_accumulate<false>(&jtile[jbase], pi, dif0, 0, c1, c2);
    } else {
        // Diagonal or ragged-edge tile: fold all constraints into one bound.
        int kmax = 63;
        if (ti == tj) kmax = min(kmax, i - (j0 + jbase));  // enforce j <= i
        kmax = min(kmax, (n - 1) - (j0 + jbase));          // enforce j <  n
        if (i >= n) kmax = -1;                             // enforce i <  n
        acc = strip_accumulate<true>(&jtile[jbase], pi, dif0, kmax, c1, c2);
    }

    // Wave reduction via WMMA, then cross-wave via LDS.
    float ws = wave_sum_wmma(acc);
    const int wid  = t >> 5;
    const int lane = t & 31;
    if (lane == 0) wsum[wid] = ws;
    __syncthreads();
    if (t == 0) {
        float total = 0.0f;
        #pragma unroll
        for (int w = 0; w < NWAVES; ++w) total += wsum[w];
        block_sums[blockIdx.x] = total;
    }
}

// Deterministic final reduction: one block folds all block sums, divides by n^2.
__global__ __launch_bounds__(NTHREADS)
void depth_loss_final_kernel(const float* __restrict__ block_sums,
                             float* __restrict__ out,
                             int nblocks, int n) {
    __shared__ float wsum[NWAVES];
    float acc = 0.0f;
    for (int idx = (int)threadIdx.x; idx < nblocks; idx += NTHREADS)
        acc += block_sums[idx];

    float ws = wave_sum_wmma(acc);
    const int wid  = (int)threadIdx.x >> 5;
    const int lane = (int)threadIdx.x & 31;
    if (lane == 0) wsum[wid] = ws;
    __syncthreads();
    if (threadIdx.x == 0) {
        float total = 0.0f;
        #pragma unroll
        for (int w = 0; w < NWAVES; ++w) total += wsum[w];
        out[0] = total / ((float)n * (float)n);
    }
}

extern "C" void kernel_launch(void* const* d_in, const int* in_sizes, int n_in,
                              void* d_out, int out_size, void* d_ws, size_t ws_size,
                              hipStream_t stream) {
    const float* p   = (const float*)d_in[0]; // predictions (N,1) f32
    const float* zsp = (const float*)d_in[1]; // z_spacing scalar
    const float* nth = (const float*)d_in[2]; // nth_slice scalar
    float*       out = (float*)d_out;         // scalar loss f32

    const int n       = in_sizes[0];
    const int T       = (n + TILE - 1) / TILE;      // tiles per dim (64 for N=8192)
    const int nblocks = T * (T + 1) / 2;            // lower-tri tiles (2080)

    float* block_sums = (float*)d_ws;               // nblocks * 4 bytes scratch

    depth_loss_tile_kernel<<<nblocks, NTHREADS, 0, stream>>>(p, zsp, nth, block_sums, n);
    depth_loss_final_kernel<<<1, NTHREADS, 0, stream>>>(block_sums, out, nblocks, n);
}